// Morph2D_59399397703755
// MI455X (gfx1250) — compile-verified
//
#include <hip/hip_runtime.h>

typedef __attribute__((ext_vector_type(2))) float v2f;
typedef __attribute__((ext_vector_type(8))) float v8f;

#define B_  2
#define H_  192
#define W_  192
#define C_  16
#define F_  8
#define HW_ (H_ * W_)      // 36864
#define KTOT 144           // 3*3*16
#define KSTEPS 36          // KTOT / 4

__global__ __launch_bounds__(256) void morph2d_wmma_f32(
    const float* __restrict__ x,   // [B,H,W,C]
    const float* __restrict__ se,  // [3,3,C,F]
    float* __restrict__ out)       // [B,F,HW] flat (== reference reshape)
{
    // Stage B matrix in LDS, transposed + zero-padded: Blds[n][p], n in [0,16), p in [0,144)
    __shared__ float Blds[16 * KTOT];
    const int tid = threadIdx.x;
    for (int idx = tid; idx < 16 * KTOT; idx += 256) {
        const int n = idx / KTOT;
        const int p = idx - n * KTOT;
        Blds[idx] = (n < F_) ? se[p * F_ + n] : 0.0f;
    }
    __syncthreads();

    const int wave = tid >> 5;
    const int lane = tid & 31;
    const int half = lane >> 4;   // 0: K pair {0,1}; 1: K pair {2,3}
    const int n    = lane & 15;   // A: row M; B/D: column N

    const int tile = blockIdx.x * 8 + wave;     // 4608 tiles total
    const int m0   = tile << 4;                 // first GEMM row of tile
    const int b    = m0 / HW_;                  // tiles never straddle batch
    const int hw0  = m0 - b * HW_;
    const int m    = hw0 + n;                   // this lane's pixel index
    const int h    = m / W_;
    const int w    = m - h * W_;

    const float* __restrict__ xb = x + (size_t)b * H_ * W_ * C_;

    v8f acc = {};
#pragma unroll
    for (int k = 0; k < KSTEPS; ++k) {
        const int j  = k >> 2;               // 3x3 tap index
        const int kh = j / 3;
        const int kw = j - kh * 3;
        const int p  = (k << 2) + (half << 1);  // patch index of A element in v0
        const int c  = p & 15;                  // channel (even -> 8B aligned)

        const int y  = h + kh - 1;
        const int xx = w + kw - 1;
        const float msk = ((unsigned)y < (unsigned)H_ && (unsigned)xx < (unsigned)W_) ? 1.0f : 0.0f;
        const int yc = min(max(y, 0), H_ - 1);
        const int xc = min(max(xx, 0), W_ - 1);

        v2f a = *(const v2f*)(xb + ((size_t)yc * W_ + xc) * C_ + c);
        a.x *= msk;
        a.y *= msk;

        v2f bv = *(const v2f*)(&Blds[n * KTOT + p]);

        // D = A(16x4) * B(4x16) + C, fp32 WMMA
        acc = __builtin_amdgcn_wmma_f32_16x16x4_f32(
            /*neg_a=*/false, a, /*neg_b=*/false, bv,
            /*c_mod=*/(short)0, acc, /*reuse_a=*/false, /*reuse_b=*/false);
    }

    // Store: lane holds D[M = r + 8*half][N = n]; only N < 8 columns are real.
    if (n < F_) {
        float* o = out + ((size_t)(b * F_ + n)) * HW_ + hw0 + (half << 3);
#pragma unroll
        for (int r = 0; r < 8; ++r) {
            o[r] = acc[r];
        }
    }
}

extern "C" void kernel_launch(void* const* d_in, const int* in_sizes, int n_in,
                              void* d_out, int out_size, void* d_ws, size_t ws_size,
                              hipStream_t stream) {
    const float* x  = (const float*)d_in[0];   // [2,192,192,16]
    const float* se = (const float*)d_in[1];   // [3,3,16,8]
    // d_in[2] (ranks) is dead code in the reference output.
    float* out = (float*)d_out;                // 2*8*36864 floats

    const int tiles  = (B_ * HW_) / 16;  // 4608
    const int blocks = tiles / 8;        // 576 blocks of 8 waves
    morph2d_wmma_f32<<<dim3(blocks), dim3(256), 0, stream>>>(x, se, out);
}